// DPMergeModule_22608707846355
// MI455X (gfx1250) — compile-verified
//
#include <hip/hip_runtime.h>
#include <cstdint>

// Problem constants (fixed by the reference's setup_inputs).
constexpr int Bd = 4, Cd = 3, Hd = 1024, Wd = 1024;

__device__ __forceinline__ void async_store_b128_from_lds(float* gdst, const float* lds_src) {
    // GLOBAL_STORE_ASYNC_FROM_LDS_B128: per-lane 16B LDS -> global DMA, tracked by ASYNCcnt.
    // Generic pointers to __shared__ carry the LDS byte offset in their low 32 bits
    uint64_t ga = (uint64_t)(uintptr_t)gdst;
    uint32_t la = (uint32_t)(uintptr_t)lds_src;
    asm volatile("global_store_async_from_lds_b128 %0, %1, off"
                 :: "v"(ga), "v"(la)
                 : "memory");
}

__device__ __forceinline__ void wait_asynccnt0() {
#if __has_builtin(__builtin_amdgcn_s_wait_asynccnt)
    __builtin_amdgcn_s_wait_asynccnt(0);
#else
    asm volatile("s_wait_asynccnt 0" ::: "memory");
#endif
}

__device__ __forceinline__ void splat(float* acc, float* cnt, float pos,
                                      float v0, float v1, float v2) {
    float x0f = floorf(pos);
    float f   = pos - x0f;          // bilinear fraction
    int   x0  = (int)x0f;
    int   x1  = x0 + 1;
    if ((unsigned)x0 < (unsigned)Wd) {
        float w0 = 1.0f - f;
        atomicAdd(&cnt[x0], w0);                 // ds_add_f32
        atomicAdd(&acc[0 * Wd + x0], v0 * w0);
        atomicAdd(&acc[1 * Wd + x0], v1 * w0);
        atomicAdd(&acc[2 * Wd + x0], v2 * w0);
    }
    if ((unsigned)x1 < (unsigned)Wd) {
        atomicAdd(&cnt[x1], f);
        atomicAdd(&acc[0 * Wd + x1], v0 * f);
        atomicAdd(&acc[1 * Wd + x1], v1 * f);
        atomicAdd(&acc[2 * Wd + x1], v2 * f);
    }
}

__global__ __launch_bounds__(256)
void dp_merge_kernel(const float* __restrict__ image,
                     const float* __restrict__ depth,
                     float* __restrict__ out) {
    // LDS: accL[3][W], cntL[W], accR[3][W], cntR[W]  = 8*1024 floats = 32 KB
    __shared__ float lds[8 * Wd];
    float* accL = lds;
    float* cntL = lds + 3 * Wd;
    float* accR = lds + 4 * Wd;
    float* cntR = lds + 7 * Wd;

    const int tid = threadIdx.x;
    const int row = blockIdx.x;          // 0 .. B*H-1, one image row per block
    const int bi  = row >> 10;           // row / H
    const int hi  = row & (Hd - 1);      // row % H

    // ---- zero accumulators (B128 LDS stores) ----
    float4  z4 = make_float4(0.f, 0.f, 0.f, 0.f);
    float4* lds4 = (float4*)lds;
    #pragma unroll
    for (int i = 0; i < (8 * Wd / 4) / 256; ++i)
        lds4[tid + i * 256] = z4;
    __syncthreads();

    // ---- load 4 pixels/thread (vectorized B128 global loads) and splat ----
    const size_t plane = (size_t)Hd * Wd;
    const float4* d4 = (const float4*)(depth + ((size_t)bi * Hd + hi) * Wd);
    const float4* c0 = (const float4*)(image + ((size_t)(bi * Cd + 0) * Hd + hi) * Wd);
    const float4* c1 = (const float4*)(image + ((size_t)(bi * Cd + 1) * Hd + hi) * Wd);
    const float4* c2 = (const float4*)(image + ((size_t)(bi * Cd + 2) * Hd + hi) * Wd);

    float4 dv = d4[tid];
    float4 p0 = c0[tid];
    float4 p1 = c1[tid];
    float4 p2 = c2[tid];

    float dArr[4]  = {dv.x, dv.y, dv.z, dv.w};
    float a0[4]    = {p0.x, p0.y, p0.z, p0.w};
    float a1[4]    = {p1.x, p1.y, p1.z, p1.w};
    float a2[4]    = {p2.x, p2.y, p2.z, p2.w};

    const int xbase = tid * 4;
    #pragma unroll
    for (int k = 0; k < 4; ++k) {
        float xf = (float)(xbase + k);
        float d  = dArr[k];
        splat(accL, cntL, xf + d, a0[k], a1[k], a2[k]);   // left view
        splat(accR, cntR, xf - d, a0[k], a1[k], a2[k]);   // right view
    }
    __syncthreads();

    // ---- normalize in place: acc /= (count == 0 ? 1 : count) ----
    #pragma unroll
    for (int i = 0; i < Wd / 256; ++i) {
        int x = tid + i * 256;
        float cl = cntL[x];
        cl = (cl == 0.0f) ? 1.0f : cl;
        accL[0 * Wd + x] = accL[0 * Wd + x] / cl;
        accL[1 * Wd + x] = accL[1 * Wd + x] / cl;
        accL[2 * Wd + x] = accL[2 * Wd + x] / cl;
        float cr = cntR[x];
        cr = (cr == 0.0f) ? 1.0f : cr;
        accR[0 * Wd + x] = accR[0 * Wd + x] / cr;
        accR[1 * Wd + x] = accR[1 * Wd + x] / cr;
        accR[2 * Wd + x] = accR[2 * Wd + x] / cr;
    }
    __syncthreads();

    // ---- write back: async DMA LDS -> global, 16B per lane per row ----
    float* oL = out + ((size_t)(bi * Cd) * Hd + hi) * Wd;          // left  (b,c,h,w)
    float* oR = oL + (size_t)Bd * Cd * plane;                      // right offset
    const int fo = tid * 4;   // 4 floats (16B) per lane covers the whole 1024-row
    #pragma unroll
    for (int ch = 0; ch < Cd; ++ch) {
        async_store_b128_from_lds(oL + (size_t)ch * plane + fo, &accL[ch * Wd + fo]);
        async_store_b128_from_lds(oR + (size_t)ch * plane + fo, &accR[ch * Wd + fo]);
    }
    wait_asynccnt0();   // drain ASYNCcnt before wave termination
}

extern "C" void kernel_launch(void* const* d_in, const int* in_sizes, int n_in,
                              void* d_out, int out_size, void* d_ws, size_t ws_size,
                              hipStream_t stream) {
    const float* image = (const float*)d_in[0];   // (4,3,1024,1024) f32
    const float* depth = (const float*)d_in[1];   // (4,1024,1024)   f32
    float* out = (float*)d_out;                   // left ++ right, 2*(4,3,1024,1024)

    dim3 grid(Bd * Hd);   // one workgroup per (b,h) image row = 4096 blocks
    dim3 block(256);      // 8 wave32 waves
    dp_merge_kernel<<<grid, block, 0, stream>>>(image, depth, out);
}